// SMFNet_23519240913301
// MI455X (gfx1250) — compile-verified
//
#include <hip/hip_runtime.h>
#include <hip/hip_bf16.h>

typedef __attribute__((ext_vector_type(2))) float v2f;
typedef __attribute__((ext_vector_type(8))) float v8f;

#define N_ROWS 1000000
#define D_IN   128
#define H_OUT  2

// ---------------------------------------------------------------------------
// Pass 1: V[i, 0:2] = X[i, :] @ Wg^T + bg   via V_WMMA_F32_16X16X4_F32
// One wave handles TWO 16-row tiles of X (two independent accumulator chains,
// shared B fragment). K=128 -> 32 wmma steps of K=4 per tile.
//
// A (16x4 f32, 2 VGPRs): lane L<16 -> row M=L, K={k,k+1}; lane L>=16 ->
//   M=L-16, K={k+2,k+3}  => per-lane 8B contiguous load at X[row*128 + k + 2*half].
// B (4x16): lane's column N = lane&15; only N in {0,1} is live (Wg rows).
// D (16x16, 8 VGPRs): VGPR j <-> row M = j + 8*half, column N = lane&15.
// ---------------------------------------------------------------------------
__global__ __launch_bounds__(256) void gemv_wmma_f32(
    const float* __restrict__ X,
    const float* __restrict__ Wg,
    const float* __restrict__ bg,
    float* __restrict__ V,      // [N_ROWS, 2] workspace
    int numPairs)
{
    const int lane = threadIdx.x & 31;
    const int wave = threadIdx.x >> 5;
    const int pair = blockIdx.x * (blockDim.x >> 5) + wave;
    if (pair >= numPairs) return;   // wave-uniform: EXEC stays all-1s below

    const int tile0 = pair * 2;
    const int half  = lane >> 4;    // 0: lanes 0-15, 1: lanes 16-31
    const int mn    = lane & 15;    // A: row-in-tile M; B/D: column N
    const bool blive = (mn < H_OUT);

    // A-operand sources for the two tiles.
    const float* arow0 = X + (size_t)(tile0 * 16 + mn) * D_IN + 2 * half;
    const float* arow1 = arow0 + 16 * D_IN;

    // B-operand: clamp row index so every lane loads a valid (cached) address;
    // dead columns are zeroed with a cndmask, no pointer compare / branch.
    const float* brow = Wg + (blive ? mn : 0) * D_IN + 2 * half;

    v8f acc0 = {};
    v8f acc1 = {};

    #pragma unroll 8
    for (int k = 0; k < D_IN; k += 4) {
        v2f b = *(const v2f*)(brow + k);
        b.x = blive ? b.x : 0.0f;
        b.y = blive ? b.y : 0.0f;
        v2f a0 = *(const v2f*)(arow0 + k);
        v2f a1 = *(const v2f*)(arow1 + k);
        acc0 = __builtin_amdgcn_wmma_f32_16x16x4_f32(
            false, a0, false, b, (short)0, acc0, false, false);
        acc1 = __builtin_amdgcn_wmma_f32_16x16x4_f32(
            false, a1, false, b, (short)0, acc1, false, false);
    }

    // Only columns 0 and 1 of D are live outputs.
    if (blive) {
        const float bias = bg[mn];
        const int mbase0 = tile0 * 16 + half * 8;
        const int mbase1 = mbase0 + 16;
        #pragma unroll
        for (int j = 0; j < 8; ++j) {
            V[(size_t)(mbase0 + j) * H_OUT + mn] = acc0[j] + bias;
            V[(size_t)(mbase1 + j) * H_OUT + mn] = acc1[j] + bias;
        }
    }
}

// ---------------------------------------------------------------------------
// Pass 2: out[i] = V[i,0] * V[(i+1)%N] + V[i,1] * V[(i+2)%N]   (both columns)
// ---------------------------------------------------------------------------
__global__ __launch_bounds__(256) void combine_roll(
    const float* __restrict__ V,
    float* __restrict__ out,
    int n)
{
    int i = blockIdx.x * blockDim.x + threadIdx.x;
    if (i >= n) return;
    int i1 = (i + 1 == n) ? 0 : i + 1;
    int i2 = (i + 2 >= n) ? (i + 2 - n) : i + 2;

    v2f f  = *(const v2f*)(V + (size_t)i  * 2);
    v2f v1 = *(const v2f*)(V + (size_t)i1 * 2);
    v2f v2 = *(const v2f*)(V + (size_t)i2 * 2);

    v2f r;
    r.x = f.x * v1.x + f.y * v2.x;
    r.y = f.x * v1.y + f.y * v2.y;
    *(v2f*)(out + (size_t)i * 2) = r;
}

// ---------------------------------------------------------------------------
// kernel_launch: inputs are (X[1e6*128] f32, Wg[2*128] f32, bg[2] f32);
// output is f32[1e6*2]. d_ws holds the intermediate V (8 MB).
// ---------------------------------------------------------------------------
extern "C" void kernel_launch(void* const* d_in, const int* in_sizes, int n_in,
                              void* d_out, int out_size, void* d_ws, size_t ws_size,
                              hipStream_t stream) {
    const float* X  = (const float*)d_in[0];
    const float* Wg = (const float*)d_in[1];
    const float* bg = (const float*)d_in[2];
    float* out = (float*)d_out;
    float* V   = (float*)d_ws;   // N_ROWS * 2 floats = 8 MB scratch

    const int numTiles = N_ROWS / 16;            // 62,500 (exact)
    const int numPairs = numTiles / 2;           // 31,250 (exact)
    const int wavesPerBlock = 256 / 32;          // 8
    const int grid1 = (numPairs + wavesPerBlock - 1) / wavesPerBlock;
    gemv_wmma_f32<<<grid1, 256, 0, stream>>>(X, Wg, bg, V, numPairs);

    const int grid2 = (N_ROWS + 255) / 256;
    combine_roll<<<grid2, 256, 0, stream>>>(V, out, N_ROWS);
}